// GAT_Encoder_15985868275835
// MI455X (gfx1250) — compile-verified
//
#include <hip/hip_runtime.h>
#include <math.h>

#define NN 50000
#define NE 800000
#define INC 128
#define HID 32
#define HEADS 4
#define NEG_SLOPE 0.2f

typedef __attribute__((ext_vector_type(2))) float v2f;
typedef __attribute__((ext_vector_type(8))) float v8f;

__device__ __forceinline__ float lrelu(float v) { return v > 0.f ? v : NEG_SLOPE * v; }

// order-preserving float atomic max (valid for non-NaN, init -inf)
__device__ __forceinline__ void atomicMaxF(float* addr, float val) {
    if (val >= 0.f) atomicMax((int*)addr, __float_as_int(val));
    else            atomicMin((unsigned int*)addr, __float_as_uint(val));
}

// ---------------- WMMA GEMM: D(MxN) = A(MxK) @ B(KxN), all row-major f32 ----
// blockDim.x = (N/16)*32, gridDim.x = M/16. Requires M%16==0, K%4==0, N%16==0.
__global__ void wmma_gemm_f32(const float* __restrict__ A, const float* __restrict__ B,
                              float* __restrict__ D, int M, int N, int K) {
    const int wave = threadIdx.x >> 5;
    const int lane = threadIdx.x & 31;
    const int half = lane >> 4;      // 0: K pair {0,1} rows 0-7 ; 1: K pair {2,3} rows 8-15
    const int l    = lane & 15;
    const int m0   = blockIdx.x << 4;
    const int n0   = wave << 4;
    if (m0 >= M) return;

    v8f acc = {0.f, 0.f, 0.f, 0.f, 0.f, 0.f, 0.f, 0.f};
    const float* ap = A + (size_t)(m0 + l) * K + 2 * half;      // A[m0+l, k+2*half .. +1]
    const float* bp = B + (size_t)(2 * half) * N + n0 + l;      // B[k+2*half, n0+l]

    for (int k = 0; k < K; k += 4) {
        v2f a;
        a.x = ap[k];
        a.y = ap[k + 1];
        v2f b;
        b.x = bp[(size_t)k * N];
        b.y = bp[(size_t)k * N + N];
        acc = __builtin_amdgcn_wmma_f32_16x16x4_f32(false, a, false, b,
                                                    (short)0, acc, false, false);
    }
#pragma unroll
    for (int i = 0; i < 8; ++i)
        D[(size_t)(m0 + i + 8 * half) * N + n0 + l] = acc[i];
}

// ---------------- fill --------------------------------------------------------
__global__ void fill_f32(float* p, float v, int n) {
    int i = blockIdx.x * blockDim.x + threadIdx.x;
    if (i < n) p[i] = v;
}

// ---------------- per-node attention logit halves -----------------------------
// h: (N, H, C) ; a_src/a_dst: (H, C) ; as/ad: (N, H)
__global__ void alpha_kernel(const float* __restrict__ h, const float* __restrict__ a_src,
                             const float* __restrict__ a_dst, float* __restrict__ as,
                             float* __restrict__ ad, int n_nodes, int H, int C) {
    int idx = blockIdx.x * blockDim.x + threadIdx.x;
    if (idx >= n_nodes * H) return;
    int hh = idx % H;
    const float* hp = h + (size_t)idx * C;
    const float* sv = a_src + hh * C;
    const float* dv = a_dst + hh * C;
    float s = 0.f, d = 0.f;
    for (int c = 0; c < C; ++c) { float v = hp[c]; s += v * sv[c]; d += v * dv[c]; }
    as[idx] = s;
    ad[idx] = d;
}

__device__ __forceinline__ void edge_sd(const int* ei, int e, int nE, int& s, int& d) {
    if (e < nE) { s = ei[e]; d = ei[nE + e]; }
    else        { s = d = e - nE; }   // self loops appended
}

// ---------------- pass 1: segment max of leaky-relu logits --------------------
__global__ void edge_max(const int* __restrict__ ei, const float* __restrict__ as,
                         const float* __restrict__ ad, float* __restrict__ m,
                         int nE, int nN, int H) {
    int e = blockIdx.x * blockDim.x + threadIdx.x;
    if (e >= nE + nN) return;
    int s, d; edge_sd(ei, e, nE, s, d);
    for (int h = 0; h < H; ++h) {
        float v = lrelu(as[s * H + h] + ad[d * H + h]);
        atomicMaxF(&m[d * H + h], v);
    }
}

// ---------------- pass 2: segment sum of exp(e - max) -------------------------
__global__ void edge_expsum(const int* __restrict__ ei, const float* __restrict__ as,
                            const float* __restrict__ ad, const float* __restrict__ m,
                            float* __restrict__ den, int nE, int nN, int H) {
    int e = blockIdx.x * blockDim.x + threadIdx.x;
    if (e >= nE + nN) return;
    int s, d; edge_sd(ei, e, nE, s, d);
    for (int h = 0; h < H; ++h) {
        float v = lrelu(as[s * H + h] + ad[d * H + h]);
        atomicAdd(&den[d * H + h], __expf(v - m[d * H + h]));
    }
}

// ---------------- pass 3: weighted scatter-aggregate --------------------------
// one 32-lane group per (edge, head); lane = channel (C == 32)
__global__ void edge_agg(const int* __restrict__ ei, const float* __restrict__ as,
                         const float* __restrict__ ad, const float* __restrict__ m,
                         const float* __restrict__ den, const float* __restrict__ h,
                         float* __restrict__ out, int nE, int nN, int H, int C) {
    int gid  = blockIdx.x * (blockDim.x >> 5) + (threadIdx.x >> 5);
    int lane = threadIdx.x & 31;
    int total = (nE + nN) * H;
    if (gid >= total) return;
    int e = gid / H, hh = gid % H;
    int s, d; edge_sd(ei, e, nE, s, d);
    float v   = lrelu(as[s * H + hh] + ad[d * H + hh]);
    float att = __expf(v - m[d * H + hh]) / den[d * H + hh];
    float hv  = h[((size_t)s * H + hh) * C + lane];
    atomicAdd(&out[((size_t)d * H + hh) * C + lane], att * hv);
}

// ---------------- finalize: +bias then ELU (in place, feeds layer 2) ----------
__global__ void bias_elu(float* __restrict__ acc, const float* __restrict__ b,
                         int n_nodes, int F) {
    int i = blockIdx.x * blockDim.x + threadIdx.x;
    if (i >= n_nodes * F) return;
    float v = acc[i] + b[i % F];
    acc[i] = v > 0.f ? v : (__expf(v) - 1.f);
}

// ---------------- finalize: +bias into output ---------------------------------
__global__ void bias_out(const float* __restrict__ acc, const float* __restrict__ b,
                         float* __restrict__ out, int n_nodes, int F) {
    int i = blockIdx.x * blockDim.x + threadIdx.x;
    if (i >= n_nodes * F) return;
    out[i] = acc[i] + b[i % F];
}

extern "C" void kernel_launch(void* const* d_in, const int* in_sizes, int n_in,
                              void* d_out, int out_size, void* d_ws, size_t ws_size,
                              hipStream_t stream) {
    const float* x      = (const float*)d_in[0];
    const int*   ei     = (const int*)d_in[1];   // (2, NE): [0]=src, [1]=dst
    const float* W1     = (const float*)d_in[2];
    const float* a_src1 = (const float*)d_in[3];
    const float* a_dst1 = (const float*)d_in[4];
    const float* b1     = (const float*)d_in[5];
    const float* W2     = (const float*)d_in[6];
    const float* a_src2 = (const float*)d_in[7];
    const float* a_dst2 = (const float*)d_in[8];
    const float* b2     = (const float*)d_in[9];
    float* out = (float*)d_out;

    // workspace carve-up (floats)
    float* ws   = (float*)d_ws;
    float* h1   = ws;                 ws += (size_t)NN * HEADS * HID;  // 6.4M
    float* acc1 = ws;                 ws += (size_t)NN * HEADS * HID;  // 6.4M
    float* as1  = ws;                 ws += (size_t)NN * HEADS;
    float* ad1  = ws;                 ws += (size_t)NN * HEADS;
    float* m1   = ws;                 ws += (size_t)NN * HEADS;
    float* den1 = ws;                 ws += (size_t)NN * HEADS;
    float* h2   = ws;                 ws += (size_t)NN * HID;
    float* acc2 = ws;                 ws += (size_t)NN * HID;
    float* as2  = ws;                 ws += (size_t)NN;
    float* ad2  = ws;                 ws += (size_t)NN;
    float* m2   = ws;                 ws += (size_t)NN;
    float* den2 = ws;                 ws += (size_t)NN;

    const int ET = NE + NN;           // edges incl. self loops
    const int TB = 256;
    auto blocks = [](int n, int t) { return (n + t - 1) / t; };

    // ---- init accumulators ----
    fill_f32<<<blocks(NN * HEADS * HID, TB), TB, 0, stream>>>(acc1, 0.f, NN * HEADS * HID);
    fill_f32<<<blocks(NN * HEADS, TB), TB, 0, stream>>>(m1, -INFINITY, NN * HEADS);
    fill_f32<<<blocks(NN * HEADS, TB), TB, 0, stream>>>(den1, 0.f, NN * HEADS);
    fill_f32<<<blocks(NN * HID, TB), TB, 0, stream>>>(acc2, 0.f, NN * HID);
    fill_f32<<<blocks(NN, TB), TB, 0, stream>>>(m2, -INFINITY, NN);
    fill_f32<<<blocks(NN, TB), TB, 0, stream>>>(den2, 0.f, NN);

    // ---- layer 1 ----
    // h1 = x @ W1 : (50000x128)@(128x128), 8 waves/block cover N=128
    wmma_gemm_f32<<<NN / 16, (INC / 16) * 32, 0, stream>>>(x, W1, h1, NN, HEADS * HID, INC);
    alpha_kernel<<<blocks(NN * HEADS, TB), TB, 0, stream>>>(h1, a_src1, a_dst1, as1, ad1,
                                                            NN, HEADS, HID);
    edge_max<<<blocks(ET, TB), TB, 0, stream>>>(ei, as1, ad1, m1, NE, NN, HEADS);
    edge_expsum<<<blocks(ET, TB), TB, 0, stream>>>(ei, as1, ad1, m1, den1, NE, NN, HEADS);
    edge_agg<<<blocks(ET * HEADS * 32, TB), TB, 0, stream>>>(ei, as1, ad1, m1, den1, h1, acc1,
                                                             NE, NN, HEADS, HID);
    bias_elu<<<blocks(NN * HEADS * HID, TB), TB, 0, stream>>>(acc1, b1, NN, HEADS * HID);

    // ---- layer 2 ----
    // h2 = elu(out1) @ W2 : (50000x128)@(128x32), 2 waves/block cover N=32
    wmma_gemm_f32<<<NN / 16, (HID / 16) * 32, 0, stream>>>(acc1, W2, h2, NN, HID, HEADS * HID);
    alpha_kernel<<<blocks(NN, TB), TB, 0, stream>>>(h2, a_src2, a_dst2, as2, ad2, NN, 1, HID);
    edge_max<<<blocks(ET, TB), TB, 0, stream>>>(ei, as2, ad2, m2, NE, NN, 1);
    edge_expsum<<<blocks(ET, TB), TB, 0, stream>>>(ei, as2, ad2, m2, den2, NE, NN, 1);
    edge_agg<<<blocks(ET * 32, TB), TB, 0, stream>>>(ei, as2, ad2, m2, den2, h2, acc2,
                                                     NE, NN, 1, HID);
    bias_out<<<blocks(NN * HID, TB), TB, 0, stream>>>(acc2, b2, out, NN, HID);
    (void)in_sizes; (void)n_in; (void)out_size; (void)ws_size;
}